// ABGNN_13022340841660
// MI455X (gfx1250) — compile-verified
//
#include <hip/hip_runtime.h>

typedef float v2f __attribute__((ext_vector_type(2)));
typedef float v8f __attribute__((ext_vector_type(8)));
typedef unsigned int v4u __attribute__((ext_vector_type(4)));
typedef int v4i __attribute__((ext_vector_type(4)));
typedef int v8i __attribute__((ext_vector_type(8)));

#define HID 128
#define IN_DIM 16
#define WPITCH 288   // k-pair row pitch in LDS floats: 256 payload + 32 pad
                     // (TDM pad_enable: +32 dwords after every 256 dwords)
                     // -> half-wave groups hit disjoint 32-bank sets

// ---------------------------------------------------------------------------
// TDM: async 2D tile load Global->LDS with padded rows.
// D# built per CDNA5 ISA 8.3/8.4:  count=1, type=2 (image), data_size=4B,
// tile == tensor (rows x 256 f32), pad 32 dwords per 256 dwords.
// clang-23 toolchain: 6-arg builtin (g0 v4u, g1 v8i, g2 v4i, g3 v4i, g4 v8i, cpol)
// ---------------------------------------------------------------------------
__device__ __forceinline__ void tdm_load_padded(unsigned lds_byte_off,
                                                const float* gsrc,
                                                unsigned rows)
{
    unsigned long long ga = (unsigned long long)(size_t)gsrc;
    const unsigned dim0 = 256;                     // elements per row
    const unsigned dim1 = rows;

    v4u g0;
    g0.x = 1u;                                     // count=1, user descriptor
    g0.y = lds_byte_off;                           // lds_addr (bytes)
    g0.z = (unsigned)ga;                           // global_addr[31:0]
    g0.w = ((unsigned)(ga >> 32) & 0x01FFFFFFu)    // global_addr[56:32]
           | (2u << 30);                           // type=2

    v8i g1;
    g1[0] = (int)((2u << 16)                       // data_size = 4 bytes
                | (1u << 20)                       // pad_enable
                | (7u << 22)                       // pad_interval: 256 dwords
                | (31u << 25));                    // pad_amount: 32 dwords
    g1[1] = (int)((dim0 & 0xFFFFu) << 16);                       // t_dim0 lo16
    g1[2] = (int)((dim0 >> 16) | ((dim1 & 0xFFFFu) << 16));      // t_dim0 hi | t_dim1 lo
    g1[3] = (int)((dim1 >> 16) | (dim0 << 16));                  // t_dim1 hi | tile_dim0
    g1[4] = (int)(dim1 & 0xFFFFu);                               // tile_dim1 (tile_dim2=0)
    g1[5] = (int)dim0;                                           // t_dim0_stride lo32
    g1[6] = 0;                                                   // stride hi16 | d1stride lo16
    g1[7] = 0;                                                   // d1stride hi32

    v4i g2 = {0, 0, 0, 0};
    v4i g3 = {0, 0, 0, 0};
    v8i g4 = {0, 0, 0, 0, 0, 0, 0, 0};
    __builtin_amdgcn_tensor_load_to_lds(g0, g1, g2, g3, g4, 0);
}

__device__ __forceinline__ unsigned lds_off_of(const void* p) {
    return (unsigned)(size_t)p;    // flat LDS aperture: addr[31:0] == LDS offset
}

// ---------------------------------------------------------------------------
// zero fill (agg / cnt scratch must be zeroed every call: harness only poisons)
// ---------------------------------------------------------------------------
__global__ void __launch_bounds__(256) zero_kernel(float* __restrict__ p, long n) {
    long i = (long)blockIdx.x * blockDim.x + threadIdx.x;
    long stride = (long)gridDim.x * blockDim.x;
    for (; i < n; i += stride) p[i] = 0.0f;
}

// ---------------------------------------------------------------------------
// One-shot pack: pair-transpose weights so a WMMA B-fragment (B[k][c],B[k+1][c])
// is one aligned 8-byte load.  Layout: Wt[kp*256 + 2c + {0,1}].
// wt layout: [Wt_self 64*256][Wt_neigh 64*256][Wt_init 8*256]
// ---------------------------------------------------------------------------
__global__ void __launch_bounds__(256) pack_weights_kernel(
    const float* __restrict__ Ws, const float* __restrict__ Wn,
    const float* __restrict__ Wi, float* __restrict__ wt)
{
    int i = blockIdx.x * blockDim.x + threadIdx.x;
    if (i < 64 * HID) {
        int kp = i >> 7, c = i & 127;
        wt[kp * 256 + 2 * c]                  = Ws[(2 * kp) * HID + c];
        wt[kp * 256 + 2 * c + 1]              = Ws[(2 * kp + 1) * HID + c];
        wt[16384 + kp * 256 + 2 * c]          = Wn[(2 * kp) * HID + c];
        wt[16384 + kp * 256 + 2 * c + 1]      = Wn[(2 * kp + 1) * HID + c];
    }
    if (i < 8 * HID) {
        int kp = i >> 7, c = i & 127;
        wt[32768 + kp * 256 + 2 * c]          = Wi[(2 * kp) * HID + c];
        wt[32768 + kp * 256 + 2 * c + 1]      = Wi[(2 * kp + 1) * HID + c];
    }
}

// ---------------------------------------------------------------------------
// Edge scatter: one wave per edge, each lane moves float4 (512B/edge bursts),
// segment-sum via global float atomics, lane 0 counts degree.
// This is the roofline-dominant stage (~2.9 GB of HBM traffic total).
// ---------------------------------------------------------------------------
__global__ void __launch_bounds__(256) scatter_kernel(
    const float* __restrict__ h, const int* __restrict__ src,
    const int* __restrict__ dst, float* __restrict__ agg,
    float* __restrict__ cnt, int E)
{
    int t = blockIdx.x * blockDim.x + threadIdx.x;
    int e = t >> 5;
    if (e >= E) return;
    int lane = t & 31;
    int s = src[e];
    int d = dst[e];
    const float4* hp = (const float4*)(h + (size_t)s * HID);
    float4 v = hp[lane];
    float* ap = agg + (size_t)d * HID + lane * 4;
    atomicAdd(ap + 0, v.x);
    atomicAdd(ap + 1, v.y);
    atomicAdd(ap + 2, v.z);
    atomicAdd(ap + 3, v.w);
    if (lane == 0) atomicAdd(cnt + d, 1.0f);
}

// ---------------------------------------------------------------------------
// h0 = relu(features @ W_init + b_init)  via V_WMMA_F32_16X16X4_F32
// W_init (pair-transposed) staged via TDM; one wave -> 16 rows x 128 cols.
// Dynamic LDS: 8 * WPITCH + 128 floats.
// ---------------------------------------------------------------------------
__global__ void __launch_bounds__(256) init_fc_kernel(
    const float* __restrict__ X, const float* __restrict__ Wt,
    const float* __restrict__ b, float* __restrict__ out, int n)
{
    extern __shared__ float smem[];
    float* ldsW = smem;                    // 8 * WPITCH
    float* ldsB = smem + 8 * WPITCH;       // 128

    const int tid = threadIdx.x;
    if (tid < HID) ldsB[tid] = b[tid];
    if ((tid >> 5) == 0) {                 // wave 0 drives the TDM
        tdm_load_padded(lds_off_of(ldsW), Wt, 8);
        __builtin_amdgcn_s_wait_tensorcnt(0);
    }
    __syncthreads();

    // force wave-uniform tile index into an SGPR
    int wv   = __builtin_amdgcn_readfirstlane((int)(threadIdx.x >> 5));
    int wave = (int)blockIdx.x * 8 + wv;
    int ntile = (n + 15) >> 4;
    if (wave >= ntile) return;             // scalar branch, EXEC untouched
    int lane = tid & 31;
    int m  = lane & 15;
    int kh = (lane >> 4) << 1;             // lanes 16..31 hold K = k0+2, k0+3
    int mh = (lane >> 4) << 3;             // D rows M+8 live in lanes 16..31
    int rowbase = wave << 4;
    int row  = rowbase + m;
    int arow = row < n ? row : 0;          // clamp loads; mask stores
    bool full = (rowbase + 16) <= n;       // scalar fast-path condition

    v2f a[4];
#pragma unroll
    for (int i = 0; i < 4; ++i)
        a[i] = *(const v2f*)(X + arow * IN_DIM + i * 4 + kh);

    for (int nt = 0; nt < 8; ++nt) {
        int col = (nt << 4) + m;
        float binit = ldsB[col];
        v8f acc;
#pragma unroll
        for (int r = 0; r < 8; ++r) acc[r] = binit;
#pragma unroll
        for (int i = 0; i < 4; ++i) {
            int kp = 2 * i + (lane >> 4);
            v2f bf = *(const v2f*)(ldsW + kp * WPITCH + 2 * col);
            acc = __builtin_amdgcn_wmma_f32_16x16x4_f32(
                false, a[i], false, bf, (short)0, acc, false, false);
        }
        if (full) {
#pragma unroll
            for (int r = 0; r < 8; ++r)
                out[(size_t)(rowbase + mh + r) * HID + col] = fmaxf(acc[r], 0.0f);
        } else {
#pragma unroll
            for (int r = 0; r < 8; ++r) {
                int orow = rowbase + mh + r;
                if (orow < n) out[(size_t)orow * HID + col] = fmaxf(acc[r], 0.0f);
            }
        }
    }
}

// ---------------------------------------------------------------------------
// SAGE combine:  out = [relu]( hprev[:n] @ Ws + (agg/max(cnt,1)) @ Wn + bs+bn )
// - Pair-transposed Ws/Wn staged by two TDM tensor loads (pad -> WPITCH rows),
//   bias pre-summed into LDS.
// - All 64 A fragments (self+neigh) preloaded to registers, mean scale folded.
// - Two independent WMMA accumulator chains (self / neigh) for ILP.
// Dynamic LDS: 2 * 64 * WPITCH + 128 floats (~148 KB of the 320 KB WGP LDS).
// ---------------------------------------------------------------------------
__global__ void __launch_bounds__(256) sage_combine_kernel(
    const float* __restrict__ hprev, const float* __restrict__ agg,
    const float* __restrict__ cnt,
    const float* __restrict__ Wts, const float* __restrict__ Wtn,
    const float* __restrict__ bs, const float* __restrict__ bn,
    float* __restrict__ out, int n, int relu)
{
    extern __shared__ float smem[];
    float* ldsWs = smem;                       // 64 * WPITCH
    float* ldsWn = smem + 64 * WPITCH;         // 64 * WPITCH
    float* ldsB  = smem + 2 * 64 * WPITCH;     // 128

    const int tid = threadIdx.x;
    if (tid < HID) ldsB[tid] = bs[tid] + bn[tid];
    if ((tid >> 5) == 0) {                     // wave 0 drives both TDM loads
        tdm_load_padded(lds_off_of(ldsWs), Wts, 64);
        tdm_load_padded(lds_off_of(ldsWn), Wtn, 64);
        __builtin_amdgcn_s_wait_tensorcnt(0);
    }
    __syncthreads();

    int wv   = __builtin_amdgcn_readfirstlane((int)(threadIdx.x >> 5));
    int wave = (int)blockIdx.x * 8 + wv;
    int ntile = (n + 15) >> 4;
    if (wave >= ntile) return;                 // scalar branch
    int lane = tid & 31;
    int m  = lane & 15;
    int kh = (lane >> 4) << 1;
    int mh = (lane >> 4) << 3;
    int rowbase = wave << 4;
    int row  = rowbase + m;
    int arow = row < n ? row : 0;
    size_t roff = (size_t)arow * HID;
    bool full = (rowbase + 16) <= n;           // scalar fast-path condition

    float c = cnt[arow];
    float scale = 1.0f / fmaxf(c, 1.0f);

    // Preload all A fragments (K=128 -> 32 frags each), scale folded in.
    v2f as[32], an[32];
#pragma unroll
    for (int i = 0; i < 32; ++i) {
        int kb = i * 4 + kh;
        as[i] = *(const v2f*)(hprev + roff + kb);
        v2f t = *(const v2f*)(agg + roff + kb);
        an[i].x = t.x * scale;
        an[i].y = t.y * scale;
    }

    for (int nt = 0; nt < 8; ++nt) {
        int col = (nt << 4) + m;
        float binit = ldsB[col];
        v8f acc_s, acc_n;
#pragma unroll
        for (int r = 0; r < 8; ++r) { acc_s[r] = binit; acc_n[r] = 0.0f; }
#pragma unroll
        for (int i = 0; i < 32; ++i) {
            int kp = 2 * i + (lane >> 4);
            v2f bsf = *(const v2f*)(ldsWs + kp * WPITCH + 2 * col);
            v2f bnf = *(const v2f*)(ldsWn + kp * WPITCH + 2 * col);
            acc_s = __builtin_amdgcn_wmma_f32_16x16x4_f32(
                false, as[i], false, bsf, (short)0, acc_s, false, false);
            acc_n = __builtin_amdgcn_wmma_f32_16x16x4_f32(
                false, an[i], false, bnf, (short)0, acc_n, false, false);
        }
        if (full) {
#pragma unroll
            for (int r = 0; r < 8; ++r) {
                float v = acc_s[r] + acc_n[r];
                if (relu) v = fmaxf(v, 0.0f);
                out[(size_t)(rowbase + mh + r) * HID + col] = v;
            }
        } else {
#pragma unroll
            for (int r = 0; r < 8; ++r) {
                int orow = rowbase + mh + r;
                if (orow < n) {
                    float v = acc_s[r] + acc_n[r];
                    if (relu) v = fmaxf(v, 0.0f);
                    out[(size_t)orow * HID + col] = v;
                }
            }
        }
    }
}

// ---------------------------------------------------------------------------
extern "C" void kernel_launch(void* const* d_in, const int* in_sizes, int n_in,
                              void* d_out, int out_size, void* d_ws, size_t ws_size,
                              hipStream_t stream)
{
    const float* features = (const float*)d_in[0];
    const float* W_init   = (const float*)d_in[1];
    const float* b_init   = (const float*)d_in[2];
    const float* W_self   = (const float*)d_in[3];
    const float* b_self   = (const float*)d_in[4];
    const float* W_neigh  = (const float*)d_in[5];
    const float* b_neigh  = (const float*)d_in[6];
    const int* src0 = (const int*)d_in[7];
    const int* dst0 = (const int*)d_in[8];
    const int* src1 = (const int*)d_in[9];
    const int* dst1 = (const int*)d_in[10];
    const int* src2 = (const int*)d_in[11];
    const int* dst2 = (const int*)d_in[12];

    const int N0 = in_sizes[0] / IN_DIM;      // 200000
    const int E0 = in_sizes[7];               // 1600000
    const int E1 = in_sizes[9];               // 800000
    const int E2 = in_sizes[11];              // 400000
    const int N1 = 100000, N2 = 50000, N3 = 25000;

    // workspace layout (floats)
    float* ws  = (float*)d_ws;
    float* h0  = ws;                               // N0*128
    float* h1  = h0 + (size_t)N0 * HID;            // N1*128
    float* h2  = h1 + (size_t)N1 * HID;            // N2*128
    float* agg = h2 + (size_t)N2 * HID;            // N1*128 (max)
    float* cnt = agg + (size_t)N1 * HID;           // N1     (max)
    float* wt  = cnt + (size_t)N1;                 // packed weights: 2*16384 + 2048
    float* wt_self = wt;
    float* wt_neigh = wt + 16384;
    float* wt_init = wt + 32768;

    auto gemm_blocks = [](int n) { int tiles = (n + 15) / 16; return (tiles + 7) / 8; };
    const size_t smem_init    = (size_t)(8 * WPITCH + HID) * sizeof(float);       // ~9.7 KB
    const size_t smem_combine = (size_t)(2 * 64 * WPITCH + HID) * sizeof(float);  // ~148 KB

    // pair-transpose the three weight matrices once
    pack_weights_kernel<<<64, 256, 0, stream>>>(W_self, W_neigh, W_init, wt);

    // h0 = relu(features @ W_init + b_init)
    init_fc_kernel<<<gemm_blocks(N0), 256, smem_init, stream>>>(
        features, wt_init, b_init, h0, N0);

    // ---- layer 0: N0 -> N1 ----
    zero_kernel<<<2048, 256, 0, stream>>>(agg, (long)N1 * HID);
    zero_kernel<<<256, 256, 0, stream>>>(cnt, (long)N1);
    scatter_kernel<<<(E0 * 32 + 255) / 256, 256, 0, stream>>>(h0, src0, dst0, agg, cnt, E0);
    sage_combine_kernel<<<gemm_blocks(N1), 256, smem_combine, stream>>>(
        h0, agg, cnt, wt_self, wt_neigh, b_self, b_neigh, h1, N1, 1);

    // ---- layer 1: N1 -> N2 ----
    zero_kernel<<<2048, 256, 0, stream>>>(agg, (long)N2 * HID);
    zero_kernel<<<256, 256, 0, stream>>>(cnt, (long)N2);
    scatter_kernel<<<(E1 * 32 + 255) / 256, 256, 0, stream>>>(h1, src1, dst1, agg, cnt, E1);
    sage_combine_kernel<<<gemm_blocks(N2), 256, smem_combine, stream>>>(
        h1, agg, cnt, wt_self, wt_neigh, b_self, b_neigh, h2, N2, 1);

    // ---- layer 2: N2 -> N3 (no relu, write to d_out) ----
    zero_kernel<<<2048, 256, 0, stream>>>(agg, (long)N3 * HID);
    zero_kernel<<<256, 256, 0, stream>>>(cnt, (long)N3);
    scatter_kernel<<<(E2 * 32 + 255) / 256, 256, 0, stream>>>(h2, src2, dst2, agg, cnt, E2);
    sage_combine_kernel<<<gemm_blocks(N3), 256, smem_combine, stream>>>(
        h2, agg, cnt, wt_self, wt_neigh, b_self, b_neigh, (float*)d_out, N3, 0);
}